// UpSample_55688545960049
// MI455X (gfx1250) — compile-verified
//
#include <hip/hip_runtime.h>

typedef __attribute__((ext_vector_type(2))) float v2f;
typedef __attribute__((ext_vector_type(8))) float v8f;

#define BB  16
#define NP  4096
#define SP  1024
#define D1  256
#define D2  512
#define CIN 768
#define C1  512
#define C2  256

__device__ __forceinline__ v8f wmma4(v2f a, v2f b, v8f c) {
  // V_WMMA_F32_16X16X4_F32: D = A(16x4) * B(4x16) + C, full fp32
  return __builtin_amdgcn_wmma_f32_16x16x4_f32(
      /*neg_a=*/false, a, /*neg_b=*/false, b,
      /*c_mod=*/(short)0, c, /*reuse_a=*/false, /*reuse_b=*/false);
}

// ---------------------------------------------------------------------------
// Kernel 1: 3-NN inverse-distance interpolation.
// One thread per query point; xyz2 for the batch staged in LDS (12 KB).
// Phase 2 loops channels so writes to interp[b][c][n] are coalesced
// (channel-major layout feeds the WMMA B-fragment loads directly).
// ---------------------------------------------------------------------------
__global__ __launch_bounds__(256) void knn_interp_kernel(
    const float* __restrict__ xyz1, const float* __restrict__ xyz2,
    const float* __restrict__ points2, float* __restrict__ interp) {
  __shared__ float sx[SP], sy[SP], sz[SP];
  const int b = blockIdx.y;
  const int n = blockIdx.x * 256 + threadIdx.x;

  const float* x2 = xyz2 + (size_t)b * 3 * SP;
  for (int i = threadIdx.x; i < SP; i += 256) {
    sx[i] = x2[i];
    sy[i] = x2[SP + i];
    sz[i] = x2[2 * SP + i];
  }
  __syncthreads();

  const float* x1 = xyz1 + (size_t)b * 3 * NP;
  const float qx = x1[n], qy = x1[NP + n], qz = x1[2 * NP + n];

  float d0 = 3.4e38f, d1v = 3.4e38f, d2v = 3.4e38f;
  int i0 = 0, i1 = 0, i2 = 0;
  for (int s = 0; s < SP; ++s) {
    const float dx = qx - sx[s], dy = qy - sy[s], dz = qz - sz[s];
    const float d = dx * dx + dy * dy + dz * dz;
    if (d < d2v) {
      if (d < d1v) {
        d2v = d1v; i2 = i1;
        if (d < d0) { d1v = d0; i1 = i0; d0 = d; i0 = s; }
        else        { d1v = d;  i1 = s; }
      } else { d2v = d; i2 = s; }
    }
  }
  d0  = fmaxf(d0, 1e-12f);
  d1v = fmaxf(d1v, 1e-12f);
  d2v = fmaxf(d2v, 1e-12f);
  const float r0 = 1.f / (d0 + 1e-8f);
  const float r1 = 1.f / (d1v + 1e-8f);
  const float r2 = 1.f / (d2v + 1e-8f);
  const float inv = 1.f / (r0 + r1 + r2);
  const float w0 = r0 * inv, w1 = r1 * inv, w2 = r2 * inv;

  const float* p2 = points2 + (size_t)b * D2 * SP;
  float* dst = interp + (size_t)b * D2 * NP + n;
  for (int c = 0; c < D2; ++c) {
    const float* row = p2 + (size_t)c * SP;  // 4KB row, cache resident
    dst[(size_t)c * NP] = w0 * row[i0] + w1 * row[i1] + w2 * row[i2];
  }
}

// ---------------------------------------------------------------------------
// GEMM1: Y1[b,o,n] = sum_c W1[o,c] * concat(points1, interp)[b,c,n] + b1[o]
// 8 waves/block; wave tile 16(M) x 64(N); K-loop step 4 via f32 WMMA.
// Fragment layouts per CDNA5 ISA:
//   A(16x4):  lane = m + 16*(k>=2), vgpr = k&1
//   B(4x16):  lane = n + 16*(k>=2), vgpr = k&1
//   C/D:      vgpr j -> rows j (lanes 0-15) and j+8 (lanes 16-31)
// K loop is split into the two concat regions so the inner loops are
// branch-free (pointer strides by 4*NP per step, no per-step select).
// ---------------------------------------------------------------------------
__global__ __launch_bounds__(256) void gemm1_kernel(
    const float* __restrict__ W, const float* __restrict__ bias,
    const float* __restrict__ p1, const float* __restrict__ interp,
    float* __restrict__ Y) {
  const int lane = threadIdx.x & 31;
  const int wave = threadIdx.x >> 5;
  const int wm = wave & 3, wn = wave >> 2;
  const int b  = blockIdx.z;
  const int o0 = blockIdx.y * 64 + wm * 16;
  const int n0 = blockIdx.x * 128 + wn * 64;
  const int l16 = lane & 15;
  const int kb  = (lane >> 4) << 1;   // 0 or 2
  const int hi8 = (lane >> 4) << 3;   // 0 or 8

  const float* wrow = W + (size_t)(o0 + l16) * CIN + kb;

  v8f acc[4];
#pragma unroll
  for (int j = 0; j < 8; ++j) {
    const float bv = bias[o0 + hi8 + j];
    acc[0][j] = bv; acc[1][j] = bv; acc[2][j] = bv; acc[3][j] = bv;
  }

  // Region 1: channels [0, 256) from points1 (branch-free inner loop)
  {
    const float* r = p1 + (size_t)b * D1 * NP + (size_t)kb * NP + n0 + l16;
    for (int c = 0; c < D1; c += 4) {
      const v2f a = *(const v2f*)(wrow + c);
#pragma unroll
      for (int t = 0; t < 4; ++t) {
        v2f bf;
        bf.x = r[t * 16];        // channel c+kb
        bf.y = r[NP + t * 16];   // channel c+kb+1
        acc[t] = wmma4(a, bf, acc[t]);
      }
      r += 4 * NP;
    }
  }
  // Region 2: channels [256, 768) from interp
  {
    const float* r = interp + (size_t)b * D2 * NP + (size_t)kb * NP + n0 + l16;
    for (int c = D1; c < CIN; c += 4) {
      const v2f a = *(const v2f*)(wrow + c);
#pragma unroll
      for (int t = 0; t < 4; ++t) {
        v2f bf;
        bf.x = r[t * 16];
        bf.y = r[NP + t * 16];
        acc[t] = wmma4(a, bf, acc[t]);
      }
      r += 4 * NP;
    }
  }

  float* yb = Y + (size_t)b * C1 * NP;
#pragma unroll
  for (int t = 0; t < 4; ++t)
#pragma unroll
    for (int j = 0; j < 8; ++j)
      yb[(size_t)(o0 + hi8 + j) * NP + n0 + t * 16 + l16] = acc[t][j];
}

// ---------------------------------------------------------------------------
// GEMM2: same structure, BN1+ReLU fused into the B-fragment load.
// ---------------------------------------------------------------------------
__global__ __launch_bounds__(256) void gemm2_kernel(
    const float* __restrict__ W, const float* __restrict__ bias,
    const float* __restrict__ X, const float* __restrict__ scale,
    const float* __restrict__ shift, float* __restrict__ Y) {
  const int lane = threadIdx.x & 31;
  const int wave = threadIdx.x >> 5;
  const int wm = wave & 3, wn = wave >> 2;
  const int b  = blockIdx.z;
  const int o0 = blockIdx.y * 64 + wm * 16;
  const int n0 = blockIdx.x * 128 + wn * 64;
  const int l16 = lane & 15;
  const int kb  = (lane >> 4) << 1;
  const int hi8 = (lane >> 4) << 3;

  const float* wrow = W + (size_t)(o0 + l16) * C1 + kb;
  const float* r    = X + (size_t)b * C1 * NP + (size_t)kb * NP + n0 + l16;
  const float* sc_p = scale + kb;
  const float* sh_p = shift + kb;

  v8f acc[4];
#pragma unroll
  for (int j = 0; j < 8; ++j) {
    const float bv = bias[o0 + hi8 + j];
    acc[0][j] = bv; acc[1][j] = bv; acc[2][j] = bv; acc[3][j] = bv;
  }

  for (int c = 0; c < C1; c += 4) {
    const v2f a = *(const v2f*)(wrow + c);
    const float s0 = sc_p[c],     h0 = sh_p[c];
    const float s1 = sc_p[c + 1], h1 = sh_p[c + 1];
#pragma unroll
    for (int t = 0; t < 4; ++t) {
      v2f bf;
      bf.x = fmaxf(0.f, s0 * r[t * 16] + h0);
      bf.y = fmaxf(0.f, s1 * r[NP + t * 16] + h1);
      acc[t] = wmma4(a, bf, acc[t]);
    }
    r += 4 * NP;
  }

  float* yb = Y + (size_t)b * C2 * NP;
#pragma unroll
  for (int t = 0; t < 4; ++t)
#pragma unroll
    for (int j = 0; j < 8; ++j)
      yb[(size_t)(o0 + hi8 + j) * NP + n0 + t * 16 + l16] = acc[t][j];
}

// ---------------------------------------------------------------------------
// BatchNorm training-mode stats per channel over (B, N); folds gamma/beta/
// mean/rstd into per-channel scale/shift.
// ---------------------------------------------------------------------------
__global__ __launch_bounds__(256) void bn_stats_kernel(
    const float* __restrict__ Y, const float* __restrict__ gamma,
    const float* __restrict__ beta, float* __restrict__ scale,
    float* __restrict__ shift, int C) {
  const int c = blockIdx.x;
  float s = 0.f, q = 0.f;
  for (int b = 0; b < BB; ++b) {
    const float* p = Y + ((size_t)b * C + c) * NP;
    for (int n = threadIdx.x; n < NP; n += 256) {
      const float v = p[n];
      s += v; q += v * v;
    }
  }
  __shared__ float rs[256], rq[256];
  rs[threadIdx.x] = s; rq[threadIdx.x] = q;
  __syncthreads();
  for (int off = 128; off > 0; off >>= 1) {
    if (threadIdx.x < off) {
      rs[threadIdx.x] += rs[threadIdx.x + off];
      rq[threadIdx.x] += rq[threadIdx.x + off];
    }
    __syncthreads();
  }
  if (threadIdx.x == 0) {
    const float cnt  = (float)(BB * NP);
    const float mean = rs[0] / cnt;
    const float var  = rq[0] / cnt - mean * mean;
    const float rstd = rsqrtf(var + 1e-5f);
    const float sc   = gamma[c] * rstd;
    scale[c] = sc;
    shift[c] = beta[c] - mean * sc;
  }
}

// Final BN2 + ReLU elementwise.
__global__ __launch_bounds__(256) void bn_relu_out_kernel(
    const float* __restrict__ Y, const float* __restrict__ scale,
    const float* __restrict__ shift, float* __restrict__ out, int total) {
  const int idx = blockIdx.x * 256 + threadIdx.x;
  if (idx >= total) return;
  const int c = (idx >> 12) & (C2 - 1);  // NP = 4096 = 2^12
  out[idx] = fmaxf(0.f, scale[c] * Y[idx] + shift[c]);
}

// ---------------------------------------------------------------------------
extern "C" void kernel_launch(void* const* d_in, const int* in_sizes, int n_in,
                              void* d_out, int out_size, void* d_ws, size_t ws_size,
                              hipStream_t stream) {
  const float* xyz1    = (const float*)d_in[0];
  const float* xyz2    = (const float*)d_in[1];
  const float* points1 = (const float*)d_in[2];
  const float* points2 = (const float*)d_in[3];
  const float* W1  = (const float*)d_in[4];
  const float* b1  = (const float*)d_in[5];
  const float* g1  = (const float*)d_in[6];
  const float* bt1 = (const float*)d_in[7];
  const float* W2  = (const float*)d_in[8];
  const float* b2  = (const float*)d_in[9];
  const float* g2  = (const float*)d_in[10];
  const float* bt2 = (const float*)d_in[11];
  float* out = (float*)d_out;

  float* ws     = (float*)d_ws;
  float* interp = ws;                               // 16*512*4096
  float* Y1     = interp + (size_t)BB * D2 * NP;    // 16*512*4096
  float* Y2     = Y1     + (size_t)BB * C1 * NP;    // 16*256*4096
  float* scale1 = Y2     + (size_t)BB * C2 * NP;
  float* shift1 = scale1 + C1;
  float* scale2 = shift1 + C1;
  float* shift2 = scale2 + C2;

  knn_interp_kernel<<<dim3(NP / 256, BB), 256, 0, stream>>>(xyz1, xyz2, points2, interp);
  gemm1_kernel<<<dim3(NP / 128, C1 / 64, BB), 256, 0, stream>>>(W1, b1, points1, interp, Y1);
  bn_stats_kernel<<<C1, 256, 0, stream>>>(Y1, g1, bt1, scale1, shift1, C1);
  gemm2_kernel<<<dim3(NP / 128, C2 / 64, BB), 256, 0, stream>>>(W2, b2, Y1, scale1, shift1, Y2);
  bn_stats_kernel<<<C2, 256, 0, stream>>>(Y2, g2, bt2, scale2, shift2, C2);
  const int total = BB * C2 * NP;
  bn_relu_out_kernel<<<(total + 255) / 256, 256, 0, stream>>>(Y2, scale2, shift2, out, total);
}